// GraphEncoder_37409165148772
// MI455X (gfx1250) — compile-verified
//
#include <hip/hip_runtime.h>
#include <math.h>

// ---------------------------------------------------------------------------
// GATv2 GraphEncoder for gfx1250 (MI455X), wave32.
//   N=50000 nodes, E=600000 edges (+N self loops), D=128, H=4, C=32, L=4.
// Pipeline per layer:
//   1) gemm2_wmma : xl = x@Wl+bl, xr = x@Wr+br   (V_WMMA_F32_16X16X4_F32)
//   2) layer_init : out=0, m=-inf, denom=0
//   3) edge_logits: per-edge head logits + segment max (atomic max trick)
//   4) edge_msg   : alpha=exp(l-m); denom += alpha; out[dst] += xl[src]*alpha
//   5) node_final : x = relu(LN(out/denom + bias + x_res))
// ---------------------------------------------------------------------------

typedef float v2f __attribute__((ext_vector_type(2)));
typedef float v8f __attribute__((ext_vector_type(8)));

#define D_DIM 128
#define NEG_SLOPE 0.2f
#define LN_EPS 1e-5f

// -------------------- float atomic max via int monotonic trick -------------
__device__ __forceinline__ void atomicMaxF(float* addr, float val) {
    if (val >= 0.0f) {
        atomicMax((int*)addr, __float_as_int(val));
    } else {
        atomicMin((unsigned int*)addr, __float_as_uint(val));
    }
}

// -------------------- 0a: 3-entry projected/normed embedding table ---------
__global__ __launch_bounds__(128)
void build_table3(const float* __restrict__ emb,     // [3,16]
                  const float* __restrict__ projW,   // [16,128]
                  const float* __restrict__ projb,   // [128]
                  const float* __restrict__ projg,   // [128]
                  const float* __restrict__ projbeta,// [128]
                  float* __restrict__ table3)        // [3,128]
{
    __shared__ float red[128];
    const int c = threadIdx.x;
    for (int t = 0; t < 3; ++t) {
        float acc = projb[c];
        #pragma unroll
        for (int j = 0; j < 16; ++j) acc += emb[t * 16 + j] * projW[j * 128 + c];
        red[c] = acc; __syncthreads();
        for (int s = 64; s > 0; s >>= 1) { if (c < s) red[c] += red[c + s]; __syncthreads(); }
        const float mu = red[0] * (1.0f / 128.0f);
        __syncthreads();
        const float dv = acc - mu;
        red[c] = dv * dv; __syncthreads();
        for (int s = 64; s > 0; s >>= 1) { if (c < s) red[c] += red[c + s]; __syncthreads(); }
        const float var = red[0] * (1.0f / 128.0f);
        __syncthreads();
        float y = dv * rsqrtf(var + LN_EPS) * projg[c] + projbeta[c];
        table3[t * 128 + c] = y > 0.0f ? y : 0.0f;
        __syncthreads();
    }
}

// -------------------- 0b: gather per-node features from table --------------
__global__ __launch_bounds__(256)
void gather_x(const int* __restrict__ node_types, const float* __restrict__ table3,
              float* __restrict__ x, int N)
{
    const int t = blockIdx.x * 256 + threadIdx.x;
    if (t >= N * 128) return;
    const int n = t >> 7;
    const int c = t & 127;
    x[t] = table3[node_types[n] * 128 + c];
}

// -------------------- 1: fused dual GEMM via f32 WMMA ----------------------
// One wave32 per 16x16 output tile; computes xl and xr tiles sharing A loads.
// A (16x4 f32) layout: lane<16 -> K{0,1} in {a.x,a.y}; lane>=16 -> K{2,3}.
// B (4x16  f32) layout: lane<16 -> K{0,1}; lane>=16 -> K{2,3}; N = lane&15.
// C/D: VGPR r -> M = r + 8*(lane>=16), N = lane&15.
__global__ __launch_bounds__(32)
void gemm2_wmma(const float* __restrict__ x,
                const float* __restrict__ Wl, const float* __restrict__ bl,
                const float* __restrict__ Wr, const float* __restrict__ br,
                float* __restrict__ xl, float* __restrict__ xr, int N)
{
    const int lane = threadIdx.x;
    const int row0 = blockIdx.x * 16;
    const int col0 = blockIdx.y * 16;
    const int half = lane >> 4;        // 0 | 1
    const int mr   = lane & 15;
    const int kb   = half * 2;         // per-lane K base within a 4-wide step

    int arow = row0 + mr; if (arow >= N) arow = N - 1;   // clamp (N%16==0 normally)
    const float* __restrict__ xrow = x + (size_t)arow * D_DIM;

    v8f accL = {};
    v8f accR = {};

    for (int k0 = 0; k0 < D_DIM; k0 += 4) {
        v2f a, bL, bR;
        a.x = xrow[k0 + kb];
        a.y = xrow[k0 + kb + 1];
        const float* wl0 = Wl + (size_t)(k0 + kb) * D_DIM + col0 + mr;
        const float* wr0 = Wr + (size_t)(k0 + kb) * D_DIM + col0 + mr;
        bL.x = wl0[0];      bL.y = wl0[D_DIM];
        bR.x = wr0[0];      bR.y = wr0[D_DIM];
        accL = __builtin_amdgcn_wmma_f32_16x16x4_f32(false, a, false, bL,
                                                     (short)0, accL, false, false);
        accR = __builtin_amdgcn_wmma_f32_16x16x4_f32(false, a, false, bR,
                                                     (short)0, accR, false, false);
    }

    const int col = col0 + mr;
    const float bLv = bl[col];
    const float bRv = br[col];
    #pragma unroll
    for (int r = 0; r < 8; ++r) {
        const int row = row0 + r + half * 8;
        if (row < N) {
            xl[(size_t)row * D_DIM + col] = accL[r] + bLv;
            xr[(size_t)row * D_DIM + col] = accR[r] + bRv;
        }
    }
}

// -------------------- 2: per-layer accumulator init ------------------------
__global__ __launch_bounds__(256)
void layer_init(float* __restrict__ outacc, float* __restrict__ m,
                float* __restrict__ denom, int N)
{
    const int t = blockIdx.x * 256 + threadIdx.x;
    if (t < N * 128) outacc[t] = 0.0f;
    if (t < N * 4) { m[t] = -__builtin_huge_valf(); denom[t] = 0.0f; }
}

// -------------------- 3: edge logits + segment max -------------------------
// One wave per edge; lane handles 4 contiguous channels; head h = lane>>3.
__global__ __launch_bounds__(256)
void edge_logits(const int* __restrict__ ei, int E, int N,
                 const float* __restrict__ xl, const float* __restrict__ xr,
                 const float* __restrict__ att,       // [4*32] this layer
                 float* __restrict__ logits,          // [Etot,4]
                 float* __restrict__ m)               // [N,4]
{
    const int e = blockIdx.x * 8 + (threadIdx.x >> 5);
    const int lane = threadIdx.x & 31;
    const int Etot = E + N;
    if (e >= Etot) return;
    int s, d;
    if (e < E) { s = ei[e]; d = ei[E + e]; } else { s = d = e - E; }

    const int c0 = lane * 4;
    const float4 xs = *(const float4*)(xl + (size_t)s * D_DIM + c0);
    const float4 xd = *(const float4*)(xr + (size_t)d * D_DIM + c0);
    const float4 av = *(const float4*)(att + c0);

    float p = 0.0f, v;
    v = xs.x + xd.x; v = v > 0.0f ? v : NEG_SLOPE * v; p += v * av.x;
    v = xs.y + xd.y; v = v > 0.0f ? v : NEG_SLOPE * v; p += v * av.y;
    v = xs.z + xd.z; v = v > 0.0f ? v : NEG_SLOPE * v; p += v * av.z;
    v = xs.w + xd.w; v = v > 0.0f ? v : NEG_SLOPE * v; p += v * av.w;

    // reduce within 8-lane head group (xor 1,2,4 stays inside the group)
    p += __shfl_xor(p, 1, 32);
    p += __shfl_xor(p, 2, 32);
    p += __shfl_xor(p, 4, 32);

    if ((lane & 7) == 0) {
        const int h = lane >> 3;
        logits[(size_t)e * 4 + h] = p;
        atomicMaxF(&m[(size_t)d * 4 + h], p);
    }
}

// -------------------- 4: alpha + message accumulation ----------------------
__global__ __launch_bounds__(256)
void edge_msg(const int* __restrict__ ei, int E, int N,
              const float* __restrict__ xl,
              const float* __restrict__ logits, const float* __restrict__ m,
              float* __restrict__ denom, float* __restrict__ outacc)
{
    const int e = blockIdx.x * 8 + (threadIdx.x >> 5);
    const int lane = threadIdx.x & 31;
    const int Etot = E + N;
    if (e >= Etot) return;
    int s, d;
    if (e < E) { s = ei[e]; d = ei[E + e]; } else { s = d = e - E; }

    const int c0 = lane * 4;
    const int h = lane >> 3;
    const float alpha = __expf(logits[(size_t)e * 4 + h] - m[(size_t)d * 4 + h]);
    if ((lane & 7) == 0) atomicAdd(&denom[(size_t)d * 4 + h], alpha);

    const float4 xs = *(const float4*)(xl + (size_t)s * D_DIM + c0);
    float* o = outacc + (size_t)d * D_DIM + c0;
    atomicAdd(o + 0, xs.x * alpha);
    atomicAdd(o + 1, xs.y * alpha);
    atomicAdd(o + 2, xs.z * alpha);
    atomicAdd(o + 3, xs.w * alpha);
}

// -------------------- 5: normalize + bias + residual + LN + ReLU -----------
// One wave per node; full-wave shuffle reductions for mean/var over 128 ch.
__global__ __launch_bounds__(256)
void node_finalize(const float* __restrict__ xin, float* __restrict__ xout,
                   const float* __restrict__ outacc, const float* __restrict__ denom,
                   const float* __restrict__ gbias, const float* __restrict__ g,
                   const float* __restrict__ b, int N)
{
    const int n = blockIdx.x * 8 + (threadIdx.x >> 5);
    const int lane = threadIdx.x & 31;
    if (n >= N) return;
    const int c0 = lane * 4;
    const int h = lane >> 3;

    const float inv = 1.0f / denom[(size_t)n * 4 + h];
    const float4 o  = *(const float4*)(outacc + (size_t)n * D_DIM + c0);
    const float4 xr = *(const float4*)(xin    + (size_t)n * D_DIM + c0);
    const float4 gb = *(const float4*)(gbias + c0);

    float v0 = o.x * inv + gb.x + xr.x;
    float v1 = o.y * inv + gb.y + xr.y;
    float v2 = o.z * inv + gb.z + xr.z;
    float v3 = o.w * inv + gb.w + xr.w;

    float sum = v0 + v1 + v2 + v3;
    #pragma unroll
    for (int off = 1; off < 32; off <<= 1) sum += __shfl_xor(sum, off, 32);
    const float mu = sum * (1.0f / 128.0f);

    const float d0 = v0 - mu, d1 = v1 - mu, d2 = v2 - mu, d3 = v3 - mu;
    float vs = d0 * d0 + d1 * d1 + d2 * d2 + d3 * d3;
    #pragma unroll
    for (int off = 1; off < 32; off <<= 1) vs += __shfl_xor(vs, off, 32);
    const float rs = rsqrtf(vs * (1.0f / 128.0f) + LN_EPS);

    const float4 gg = *(const float4*)(g + c0);
    const float4 bb = *(const float4*)(b + c0);
    float4 y;
    y.x = fmaxf(d0 * rs * gg.x + bb.x, 0.0f);
    y.y = fmaxf(d1 * rs * gg.y + bb.y, 0.0f);
    y.z = fmaxf(d2 * rs * gg.z + bb.z, 0.0f);
    y.w = fmaxf(d3 * rs * gg.w + bb.w, 0.0f);
    *(float4*)(xout + (size_t)n * D_DIM + c0) = y;
}

// ---------------------------------------------------------------------------
extern "C" void kernel_launch(void* const* d_in, const int* in_sizes, int n_in,
                              void* d_out, int out_size, void* d_ws, size_t ws_size,
                              hipStream_t stream) {
    const int*   node_types = (const int*)d_in[0];
    const int*   edge_index = (const int*)d_in[1];
    const float* emb        = (const float*)d_in[2];
    const float* projW      = (const float*)d_in[3];
    const float* projb      = (const float*)d_in[4];
    const float* projg      = (const float*)d_in[5];
    const float* projbeta   = (const float*)d_in[6];
    const float* Wl         = (const float*)d_in[7];
    const float* bl         = (const float*)d_in[8];
    const float* Wr         = (const float*)d_in[9];
    const float* br         = (const float*)d_in[10];
    const float* att        = (const float*)d_in[11];
    const float* gat_bias   = (const float*)d_in[12];
    const float* ln_g       = (const float*)d_in[13];
    const float* ln_b       = (const float*)d_in[14];

    const int N    = in_sizes[0];
    const int E    = in_sizes[1] / 2;
    const int L    = in_sizes[7] / (D_DIM * D_DIM);
    const int Etot = E + N;

    // workspace carve-up (floats)
    float* ws = (float*)d_ws;
    size_t off = 0;
    float* x      = ws + off; off += (size_t)N * D_DIM;
    float* xl     = ws + off; off += (size_t)N * D_DIM;
    float* xr     = ws + off; off += (size_t)N * D_DIM;
    float* outacc = ws + off; off += (size_t)N * D_DIM;
    float* mbuf   = ws + off; off += (size_t)N * 4;
    float* denom  = ws + off; off += (size_t)N * 4;
    float* logits = ws + off; off += (size_t)Etot * 4;
    float* table3 = ws + off; off += 3 * D_DIM;

    // input projection: 3-row table, then gather
    build_table3<<<1, 128, 0, stream>>>(emb, projW, projb, projg, projbeta, table3);
    gather_x<<<(N * 128 + 255) / 256, 256, 0, stream>>>(node_types, table3, x, N);

    const int edgeBlocks = (Etot + 7) / 8;
    const int nodeBlocks = (N + 7) / 8;
    const int initBlocks = (N * 128 + 255) / 256;

    for (int l = 0; l < L; ++l) {
        dim3 gg((N + 15) / 16, D_DIM / 16);
        gemm2_wmma<<<gg, 32, 0, stream>>>(x,
                                          Wl + (size_t)l * D_DIM * D_DIM, bl + (size_t)l * D_DIM,
                                          Wr + (size_t)l * D_DIM * D_DIM, br + (size_t)l * D_DIM,
                                          xl, xr, N);
        layer_init<<<initBlocks, 256, 0, stream>>>(outacc, mbuf, denom, N);
        edge_logits<<<edgeBlocks, 256, 0, stream>>>(edge_index, E, N, xl, xr,
                                                    att + (size_t)l * D_DIM, logits, mbuf);
        edge_msg<<<edgeBlocks, 256, 0, stream>>>(edge_index, E, N, xl, logits, mbuf,
                                                 denom, outacc);
        float* xout = (l == L - 1) ? (float*)d_out : x;
        node_finalize<<<nodeBlocks, 256, 0, stream>>>(x, xout, outacc, denom,
                                                      gat_bias + (size_t)l * D_DIM,
                                                      ln_g + (size_t)l * D_DIM,
                                                      ln_b + (size_t)l * D_DIM, N);
    }
    (void)n_in; (void)out_size; (void)ws_size;
}